// ModelTorch_65335042507144
// MI455X (gfx1250) — compile-verified
//
#include <hip/hip_runtime.h>
#include <stdint.h>

// Problem constants from the reference (row stride of req_to_token).
#define MAX_CONTEXT 32768

// Native clang vector so __builtin_nontemporal_* accepts it (HIP's int4 is a
// class and gets rejected). Lowers to b128 global ops.
typedef int v4i __attribute__((ext_vector_type(4)));

// One grid.x column per request segment; grid.y slices the segment across
// multiple blocks so all WGPs get work (only BATCH=64 segments exist).
// wave32: 256 threads = 8 waves per block.
__global__ __launch_bounds__(256) void ragged_kv_copy_kernel(
    const int* __restrict__ req_to_token,      // [POOL_SIZE, MAX_CONTEXT]
    const int* __restrict__ req_pool_indices,  // [BATCH]
    const int* __restrict__ chunk_starts,      // [BATCH]
    const int* __restrict__ chunk_cu_seq_lens, // [BATCH+1]
    int* __restrict__ out)                     // [T]
{
    const int seg = blockIdx.x;

    // Per-segment metadata (scalar loads, SMEM path / KMcnt).
    const int t0  = chunk_cu_seq_lens[seg];
    const int t1  = chunk_cu_seq_lens[seg + 1];
    const int len = t1 - t0;
    if (len <= 0) return;

    const int* __restrict__ src = req_to_token
        + (size_t)req_pool_indices[seg] * (size_t)MAX_CONTEXT
        + (size_t)chunk_starts[seg];
    int* __restrict__ dst = out + t0;

    const int tid    = (int)(blockIdx.y * blockDim.x + threadIdx.x);
    const int stride = (int)(gridDim.y * blockDim.x);

    // gfx1250 global_prefetch_b8 on first line this thread touches.
    if (tid < len) __builtin_prefetch(src + tid, 0, 1);

    if ((((uintptr_t)src ^ (uintptr_t)dst) & 15u) == 0) {
        // Source and destination share 16B phase: peel to alignment, then
        // stream b128 non-temporal (data is touched exactly once).
        int head = (int)((-(intptr_t)dst) & 15) >> 2;   // ints until dst is 16B aligned
        if (head > len) head = len;
        if (tid < head) dst[tid] = src[tid];

        const int body = (len - head) >> 2;             // number of 4-int groups
        const v4i* __restrict__ s4 = (const v4i*)(src + head);
        v4i* __restrict__       d4 = (v4i*)(dst + head);
        for (int j = tid; j < body; j += stride) {
            v4i v = __builtin_nontemporal_load(s4 + j);    // global_load_b128 th:NT
            __builtin_nontemporal_store(v, d4 + j);        // global_store_b128 th:NT
        }

        const int tail0 = head + (body << 2);
        for (int j = tail0 + tid; j < len; j += stride) {
            dst[j] = src[j];
        }
    } else {
        // Mismatched phase: scalar b32 stream (still ~45ns of HBM time total).
        for (int j = tid; j < len; j += stride) {
            int v = __builtin_nontemporal_load(src + j);
            __builtin_nontemporal_store(v, dst + j);
        }
    }
}

extern "C" void kernel_launch(void* const* d_in, const int* in_sizes, int n_in,
                              void* d_out, int out_size, void* d_ws, size_t ws_size,
                              hipStream_t stream) {
    (void)n_in; (void)d_ws; (void)ws_size; (void)out_size;

    // setup_inputs() order:
    //   0: req_to_token      int32 [POOL_SIZE*MAX_CONTEXT]
    //   1: req_pool_indices  int32 [BATCH]
    //   2: chunk_starts      int32 [BATCH]
    //   3: chunk_seq_lens    int32 [BATCH]          (implied by cu_seq_lens; unused)
    //   4: chunk_cu_seq_lens int32 [BATCH+1]
    //   5: num_chunk_tokens  int32 [1]              (== out_size; unused on device)
    const int* req_to_token      = (const int*)d_in[0];
    const int* req_pool_indices  = (const int*)d_in[1];
    const int* chunk_starts      = (const int*)d_in[2];
    const int* chunk_cu_seq_lens = (const int*)d_in[4];
    int* out = (int*)d_out;   // int32 payload; element size matches harness view

    const int batch = in_sizes[1];

    // 64 x 8 = 512 blocks of 8 waves: enough parallelism to cover latency on
    // every WGP; each thread moves at most 1 v4i group (len <= 4096).
    dim3 grid((unsigned)batch, 8, 1);
    ragged_kv_copy_kernel<<<grid, 256, 0, stream>>>(
        req_to_token, req_pool_indices, chunk_starts, chunk_cu_seq_lens, out);
}